// SAModuleSlow_26834955666007
// MI455X (gfx1250) — compile-verified
//
#include <hip/hip_runtime.h>
#include <float.h>

typedef float v2f __attribute__((ext_vector_type(2)));
typedef float v8f __attribute__((ext_vector_type(8)));

#define NP_   8192
#define B_    2
#define C_    64
#define M_    4096
#define K_    128
#define H_    128
#define CK_   68          // C+3 padded to multiple of 4
#define R2_   0.04f
#define CAND_ 2048
#define H1S_  (H_ + 4)    // padded LDS stride (bank-conflict free)
#define W1P_  (CK_ / 2)   // 34 k-pairs for layer 1
#define W2P_  (H_ / 2)    // 64 k-pairs for layer 2

// ---------------------------------------------------------------------------
// Kernel 1: farthest point sampling, one block per cloud.
// ---------------------------------------------------------------------------
__global__ __launch_bounds__(1024) void k_fps(const float* __restrict__ pos,
                                              int* __restrict__ fpsIdx) {
  const int b = blockIdx.x;
  const int tid = threadIdx.x;
  const int lane = tid & 31;
  const int wid = tid >> 5;
  const int base = b * NP_;

  float px[8], py[8], pz[8], md[8];
#pragma unroll
  for (int s = 0; s < 8; ++s) {
    const int p = tid + (s << 10);
    px[s] = pos[(base + p) * 3 + 0];
    py[s] = pos[(base + p) * 3 + 1];
    pz[s] = pos[(base + p) * 3 + 2];
    md[s] = FLT_MAX;
  }

  __shared__ float wbv[32];
  __shared__ int wbi[32];
  __shared__ int winner;

  int far = 0;
  for (int i = 0; i < M_; ++i) {
    if (tid == 0) fpsIdx[b * M_ + i] = far;
    const float fx = pos[(base + far) * 3 + 0];
    const float fy = pos[(base + far) * 3 + 1];
    const float fz = pos[(base + far) * 3 + 2];

    float best = -1.0f;
    int bidx = 0;
#pragma unroll
    for (int s = 0; s < 8; ++s) {
      const float dx = __fsub_rn(px[s], fx);
      const float dy = __fsub_rn(py[s], fy);
      const float dz = __fsub_rn(pz[s], fz);
      float d = __fmul_rn(dx, dx);
      d = __fadd_rn(d, __fmul_rn(dy, dy));
      d = __fadd_rn(d, __fmul_rn(dz, dz));
      const float m = md[s] < d ? md[s] : d;
      md[s] = m;
      if (m > best) { best = m; bidx = tid + (s << 10); }   // ascending idx order
    }
#pragma unroll
    for (int off = 16; off > 0; off >>= 1) {
      const float ov = __shfl_xor(best, off, 32);
      const int oi = __shfl_xor(bidx, off, 32);
      if (ov > best || (ov == best && oi < bidx)) { best = ov; bidx = oi; }
    }
    if (lane == 0) { wbv[wid] = best; wbi[wid] = bidx; }
    __syncthreads();
    if (wid == 0) {
      float v = wbv[lane];
      int ix = wbi[lane];
#pragma unroll
      for (int off = 16; off > 0; off >>= 1) {
        const float ov = __shfl_xor(v, off, 32);
        const int oi = __shfl_xor(ix, off, 32);
        if (ov > v || (ov == v && oi < ix)) { v = ov; ix = oi; }
      }
      if (lane == 0) winner = ix;
    }
    __syncthreads();
    far = winner;
  }
}

// ---------------------------------------------------------------------------
// Kernel 2: bitonic sort of the M sampled indices per cloud; emit qpos/batch.
// ---------------------------------------------------------------------------
__global__ __launch_bounds__(1024) void k_sort(const float* __restrict__ pos,
                                               int* __restrict__ fpsIdx,
                                               float* __restrict__ qpos,
                                               int* __restrict__ batchOut) {
  __shared__ int s[M_];
  const int b = blockIdx.x;
  const int tid = threadIdx.x;
  for (int i = tid; i < M_; i += 1024) s[i] = fpsIdx[b * M_ + i];
  __syncthreads();
  for (int k = 2; k <= M_; k <<= 1) {
    for (int j = k >> 1; j > 0; j >>= 1) {
      for (int i = tid; i < M_; i += 1024) {
        const int ij = i ^ j;
        if (ij > i) {
          const int a = s[i], c = s[ij];
          const bool up = ((i & k) == 0);
          if ((a > c) == up) { s[i] = c; s[ij] = a; }
        }
      }
      __syncthreads();
    }
  }
  for (int m = tid; m < M_; m += 1024) {
    const int j = s[m];
    const int g = b * M_ + m;
    fpsIdx[g] = j;
    qpos[g * 3 + 0] = pos[(b * NP_ + j) * 3 + 0];
    qpos[g * 3 + 1] = pos[(b * NP_ + j) * 3 + 1];
    qpos[g * 3 + 2] = pos[(b * NP_ + j) * 3 + 2];
    batchOut[g] = b;
  }
}

// ---------------------------------------------------------------------------
// Kernel 3: radius + K-nearest selection, one block per center.
// valid set == { p : d2(p) <= min(R^2, Kth smallest d2) }
// ---------------------------------------------------------------------------
__global__ __launch_bounds__(256) void k_radius(const float* __restrict__ pos,
                                                const int* __restrict__ fpsIdx,
                                                int* __restrict__ nbr,
                                                int* __restrict__ nbrCnt) {
  __shared__ float cd[CAND_];
  __shared__ int ci[CAND_];
  __shared__ int cnt;
  const int g = blockIdx.x;
  const int b = g / M_;
  const int tid = threadIdx.x;
  if (tid == 0) cnt = 0;
  __syncthreads();
  const int j0 = fpsIdx[g];
  const float qx = pos[(b * NP_ + j0) * 3 + 0];
  const float qy = pos[(b * NP_ + j0) * 3 + 1];
  const float qz = pos[(b * NP_ + j0) * 3 + 2];
  for (int p = tid; p < NP_; p += 256) {
    const float dx = __fsub_rn(pos[(b * NP_ + p) * 3 + 0], qx);
    const float dy = __fsub_rn(pos[(b * NP_ + p) * 3 + 1], qy);
    const float dz = __fsub_rn(pos[(b * NP_ + p) * 3 + 2], qz);
    float d = __fmul_rn(dx, dx);
    d = __fadd_rn(d, __fmul_rn(dy, dy));
    d = __fadd_rn(d, __fmul_rn(dz, dz));
    if (d <= R2_) {
      const int slot = atomicAdd(&cnt, 1);
      if (slot < CAND_) { cd[slot] = d; ci[slot] = p; }
    }
  }
  __syncthreads();
  const int n = cnt < CAND_ ? cnt : CAND_;
  if (n <= K_) {
    for (int t = tid; t < n; t += 256) nbr[g * K_ + t] = ci[t];
    if (tid == 0) nbrCnt[g] = n;
  } else {
    for (int t = n + tid; t < CAND_; t += 256) cd[t] = FLT_MAX;
    __syncthreads();
    for (int k = 2; k <= CAND_; k <<= 1) {
      for (int j = k >> 1; j > 0; j >>= 1) {
        for (int i = tid; i < CAND_; i += 256) {
          const int ij = i ^ j;
          if (ij > i) {
            const float a = cd[i], c = cd[ij];
            const bool up = ((i & k) == 0);
            if ((a > c) == up) {
              cd[i] = c; cd[ij] = a;
              const int t0 = ci[i]; ci[i] = ci[ij]; ci[ij] = t0;
            }
          }
        }
        __syncthreads();
      }
    }
    for (int t = tid; t < K_; t += 256) nbr[g * K_ + t] = ci[t];
    if (tid == 0) nbrCnt[g] = K_;
  }
}

// ---------------------------------------------------------------------------
// Kernel 4: repack weights into pair-interleaved B-fragment layout:
//   Wpair[k/2][n][2] = { W[k][n], W[k+1][n] }   (W1 zero-padded to 68 rows)
// so each WMMA B-fragment is ONE aligned global_load_b64, 128B/half-wave.
// ---------------------------------------------------------------------------
__global__ void k_prepw(const float* __restrict__ W1, const float* __restrict__ W2,
                        float* __restrict__ W1pair, float* __restrict__ W2pair) {
  const int i = blockIdx.x * 256 + threadIdx.x;
  const int n1 = W1P_ * H_;   // 4352
  const int n2 = W2P_ * H_;   // 8192
  if (i < n1) {
    const int kp = i / H_, n = i % H_;
    const int k = kp * 2;
    W1pair[(size_t)kp * (2 * H_) + n * 2 + 0] = (k     < C_ + 3) ? W1[k * H_ + n] : 0.0f;
    W1pair[(size_t)kp * (2 * H_) + n * 2 + 1] = (k + 1 < C_ + 3) ? W1[(k + 1) * H_ + n] : 0.0f;
  } else if (i < n1 + n2) {
    const int t = i - n1;
    const int kp = t / H_, n = t % H_;
    W2pair[(size_t)kp * (2 * H_) + n * 2 + 0] = W2[(kp * 2) * H_ + n];
    W2pair[(size_t)kp * (2 * H_) + n * 2 + 1] = W2[(kp * 2 + 1) * H_ + n];
  }
}

// ---------------------------------------------------------------------------
// Kernel 5: fused PointNetConv MLP + max-aggregation via V_WMMA_F32_16X16X4_F32.
// One block (8 waves) per center; wave w owns output columns [16w,16w+16) for
// both GEMMs so the row-max reduction stays in-wave. Rows done in 2 halves of
// 64 to fit LDS. k-loop outermost: one B-fragment (global_load_b64) feeds 4
// row-tile accumulators -> 4x less W traffic per WMMA.
// ---------------------------------------------------------------------------
__global__ __launch_bounds__(256) void k_mlp(
    const float* __restrict__ x, const float* __restrict__ pos,
    const float* __restrict__ W1pair, const float* __restrict__ b1,
    const float* __restrict__ W2pair, const float* __restrict__ b2,
    const int* __restrict__ fpsIdx, const int* __restrict__ nbr,
    const int* __restrict__ nbrCnt, float* __restrict__ out) {
  __shared__ float msg[64][CK_];   // stride 68  -> banks 4*m apart, conflict-free
  __shared__ float h1s[64][H1S_];  // stride 132 -> banks 4*m apart, conflict-free

  const int g = blockIdx.x;
  const int b = g / M_;
  const int tid = threadIdx.x;
  const int lane = tid & 31;
  const int w = tid >> 5;
  const int cnt = nbrCnt[g];
  const int j0 = fpsIdx[g];
  const float qx = pos[(b * NP_ + j0) * 3 + 0];
  const float qy = pos[(b * NP_ + j0) * 3 + 1];
  const float qz = pos[(b * NP_ + j0) * 3 + 2];

  const int ml = lane & 15;
  const int nc = (w << 4) + ml;            // output column owned by this lane
  const int hi = lane >> 4;                // which K-half of the wave
  const float bias1 = b1[nc];
  const float bias2 = b2[nc];
  float colmax = -FLT_MAX;

  for (int hf = 0; hf < 2; ++hf) {  // two row-halves of 64 neighbors
    // ---- build msg[0..63][0..67] : [x_j | pos_j - pos_i | 0] ----
    {
      const int r = tid >> 2;           // 4 threads per row
      const int q = tid & 3;
      const int rg = (hf << 6) + r;     // global neighbor row
      const int cb = q * 17;
      if (rg < cnt) {
        const int j = nbr[g * K_ + rg];
        const float* xr = x + (size_t)(b * NP_ + j) * C_;
        const float rx = __fsub_rn(pos[(b * NP_ + j) * 3 + 0], qx);
        const float ry = __fsub_rn(pos[(b * NP_ + j) * 3 + 1], qy);
        const float rz = __fsub_rn(pos[(b * NP_ + j) * 3 + 2], qz);
        for (int c = cb; c < cb + 17; ++c) {
          float v;
          if (c < C_)       v = xr[c];
          else if (c == 64) v = rx;
          else if (c == 65) v = ry;
          else if (c == 66) v = rz;
          else              v = 0.0f;
          msg[r][c] = v;
        }
      } else {
        for (int c = cb; c < cb + 17; ++c) msg[r][c] = 0.0f;
      }
    }
    __syncthreads();

    // ---- GEMM1: h1 = relu(msg @ W1 + b1); 1 B-frag -> 4 row tiles ----
    {
      v8f a0 = {}, a1 = {}, a2 = {}, a3 = {};
#pragma unroll 2
      for (int k0 = 0; k0 < CK_; k0 += 4) {
        const int kb = k0 + (hi << 1);
        const float2 bb =
            *(const float2*)(W1pair + (size_t)((k0 >> 1) + hi) * (2 * H_) + nc * 2);
        v2f bv; bv.x = bb.x; bv.y = bb.y;
        float2 t; v2f av;
        t = *(const float2*)&msg[ml][kb];      av.x = t.x; av.y = t.y;
        a0 = __builtin_amdgcn_wmma_f32_16x16x4_f32(false, av, false, bv, (short)0, a0, false, false);
        t = *(const float2*)&msg[16 + ml][kb]; av.x = t.x; av.y = t.y;
        a1 = __builtin_amdgcn_wmma_f32_16x16x4_f32(false, av, false, bv, (short)0, a1, false, false);
        t = *(const float2*)&msg[32 + ml][kb]; av.x = t.x; av.y = t.y;
        a2 = __builtin_amdgcn_wmma_f32_16x16x4_f32(false, av, false, bv, (short)0, a2, false, false);
        t = *(const float2*)&msg[48 + ml][kb]; av.x = t.x; av.y = t.y;
        a3 = __builtin_amdgcn_wmma_f32_16x16x4_f32(false, av, false, bv, (short)0, a3, false, false);
      }
#pragma unroll
      for (int v = 0; v < 8; ++v) {
        const int rr = v + (hi << 3);
        float t;
        t = a0[v] + bias1; h1s[ 0 + rr][nc] = t > 0.0f ? t : 0.0f;
        t = a1[v] + bias1; h1s[16 + rr][nc] = t > 0.0f ? t : 0.0f;
        t = a2[v] + bias1; h1s[32 + rr][nc] = t > 0.0f ? t : 0.0f;
        t = a3[v] + bias1; h1s[48 + rr][nc] = t > 0.0f ? t : 0.0f;
      }
    }
    __syncthreads();

    // ---- GEMM2: h2 = relu(h1 @ W2 + b2); masked running max over rows ----
    {
      v8f a0 = {}, a1 = {}, a2 = {}, a3 = {};
#pragma unroll 2
      for (int k0 = 0; k0 < H_; k0 += 4) {
        const int kb = k0 + (hi << 1);
        const float2 bb =
            *(const float2*)(W2pair + (size_t)((k0 >> 1) + hi) * (2 * H_) + nc * 2);
        v2f bv; bv.x = bb.x; bv.y = bb.y;
        float2 t; v2f av;
        t = *(const float2*)&h1s[ml][kb];      av.x = t.x; av.y = t.y;
        a0 = __builtin_amdgcn_wmma_f32_16x16x4_f32(false, av, false, bv, (short)0, a0, false, false);
        t = *(const float2*)&h1s[16 + ml][kb]; av.x = t.x; av.y = t.y;
        a1 = __builtin_amdgcn_wmma_f32_16x16x4_f32(false, av, false, bv, (short)0, a1, false, false);
        t = *(const float2*)&h1s[32 + ml][kb]; av.x = t.x; av.y = t.y;
        a2 = __builtin_amdgcn_wmma_f32_16x16x4_f32(false, av, false, bv, (short)0, a2, false, false);
        t = *(const float2*)&h1s[48 + ml][kb]; av.x = t.x; av.y = t.y;
        a3 = __builtin_amdgcn_wmma_f32_16x16x4_f32(false, av, false, bv, (short)0, a3, false, false);
      }
#pragma unroll
      for (int v = 0; v < 8; ++v) {
        const int rowb = (hf << 6) + v + (hi << 3);
        float t;
        t = a0[v] + bias2; t = t > 0.0f ? t : 0.0f;
        colmax = (rowb +  0 < cnt && t > colmax) ? t : colmax;   // selects only
        t = a1[v] + bias2; t = t > 0.0f ? t : 0.0f;
        colmax = (rowb + 16 < cnt && t > colmax) ? t : colmax;
        t = a2[v] + bias2; t = t > 0.0f ? t : 0.0f;
        colmax = (rowb + 32 < cnt && t > colmax) ? t : colmax;
        t = a3[v] + bias2; t = t > 0.0f ? t : 0.0f;
        colmax = (rowb + 48 < cnt && t > colmax) ? t : colmax;
      }
    }
    __syncthreads();
  }

  // combine the two half-wave copies of each column (rows v vs v+8)
  const float o = __shfl_xor(colmax, 16, 32);
  colmax = o > colmax ? o : colmax;
  if (hi == 0) out[(size_t)g * H_ + nc] = colmax;
}

// ---------------------------------------------------------------------------
extern "C" void kernel_launch(void* const* d_in, const int* in_sizes, int n_in,
                              void* d_out, int out_size, void* d_ws, size_t ws_size,
                              hipStream_t stream) {
  (void)in_sizes; (void)n_in; (void)out_size; (void)ws_size;
  const float* x   = (const float*)d_in[0];
  const float* pos = (const float*)d_in[1];
  // d_in[2] = batch (layout known: b = index / NP_), unused
  const float* W1  = (const float*)d_in[3];
  const float* b1  = (const float*)d_in[4];
  const float* W2  = (const float*)d_in[5];
  const float* b2  = (const float*)d_in[6];

  float* out      = (float*)d_out;                       // [B*M, H]
  float* qpos     = out + (size_t)B_ * M_ * H_;          // [B*M, 3]
  int*   batchOut = (int*)(qpos + (size_t)B_ * M_ * 3);  // [B*M]

  char* ws = (char*)d_ws;
  const size_t nFps = (size_t)B_ * M_;
  const size_t nNbr = (size_t)B_ * M_ * K_;
  int*   fpsIdx = (int*)ws;
  int*   nbr    = (int*)(ws + sizeof(int) * nFps);
  int*   nbrCnt = (int*)(ws + sizeof(int) * (nFps + nNbr));
  float* W1pair = (float*)(ws + sizeof(int) * (nFps + nNbr + nFps));        // 8B aligned
  float* W2pair = W1pair + (size_t)W1P_ * 2 * H_;

  k_fps   <<<B_, 1024, 0, stream>>>(pos, fpsIdx);
  k_sort  <<<B_, 1024, 0, stream>>>(pos, fpsIdx, qpos, batchOut);
  k_prepw <<<((W1P_ + W2P_) * H_ + 255) / 256, 256, 0, stream>>>(W1, W2, W1pair, W2pair);
  k_radius<<<B_ * M_, 256, 0, stream>>>(pos, fpsIdx, nbr, nbrCnt);
  k_mlp   <<<B_ * M_, 256, 0, stream>>>(x, pos, W1pair, b1, W2pair, b2,
                                        fpsIdx, nbr, nbrCnt, out);
}